// BaseLayer_67705864454265
// MI455X (gfx1250) — compile-verified
//
#include <hip/hip_runtime.h>

typedef float v2f __attribute__((ext_vector_type(2)));
typedef float v4f __attribute__((ext_vector_type(4)));
typedef float v8f __attribute__((ext_vector_type(8)));

#define NEXP 4
#define MDIM 1024
#define ODIM 1024

// ---------------- init kernels ----------------
__global__ void zero_out_kernel(float* __restrict__ out, int n4) {
    int i = blockIdx.x * blockDim.x + threadIdx.x;     // n4 = n/4, out 16B aligned
    if (i < n4) ((v4f*)out)[i] = (v4f){0.f, 0.f, 0.f, 0.f};
}

__global__ void init_ws_kernel(int* __restrict__ token_for_slot,
                               float* __restrict__ prob_sum,
                               int* __restrict__ cnt, int nslots) {
    int i = blockIdx.x * blockDim.x + threadIdx.x;
    if (i < nslots) token_for_slot[i] = -1;
    if (i < NEXP) { prob_sum[i] = 0.f; cnt[i] = 0; }
}

// ---------------- gate: logits -> softmax -> top1 ----------------
// one wave (32 lanes) per token; 8 tokens per 256-thread block
__global__ void gate_kernel(const float* __restrict__ features,
                            const float* __restrict__ gate_w,
                            int* __restrict__ expert_idx,
                            float* __restrict__ gate_val,
                            float* __restrict__ prob_sum,
                            int* __restrict__ cnt, int S) {
    __shared__ float sProb[NEXP];
    __shared__ int   sCnt[NEXP];
    int tid = threadIdx.x;
    if (tid < NEXP) { sProb[tid] = 0.f; sCnt[tid] = 0; }
    __syncthreads();

    int wave = tid >> 5;
    int lane = tid & 31;
    int s = blockIdx.x * 8 + wave;
    if (s < S) {
        const float* f = features + (size_t)s * MDIM;
        float a0 = 0.f, a1 = 0.f, a2 = 0.f, a3 = 0.f;
        for (int m = lane; m < MDIM; m += 32) {
            float x = f[m];
            const float* g = gate_w + m * NEXP;
            a0 += x * g[0]; a1 += x * g[1]; a2 += x * g[2]; a3 += x * g[3];
        }
        for (int off = 16; off > 0; off >>= 1) {
            a0 += __shfl_xor(a0, off, 32);
            a1 += __shfl_xor(a1, off, 32);
            a2 += __shfl_xor(a2, off, 32);
            a3 += __shfl_xor(a3, off, 32);
        }
        if (lane == 0) {
            float lg[NEXP] = {a0, a1, a2, a3};
            float mx = lg[0];
            for (int e = 1; e < NEXP; e++) mx = fmaxf(mx, lg[e]);
            float ex[NEXP]; float sum = 0.f;
            for (int e = 0; e < NEXP; e++) { ex[e] = __expf(lg[e] - mx); sum += ex[e]; }
            float inv = 1.f / sum;
            int sel = 0; float best = ex[0] * inv;
            for (int e = 0; e < NEXP; e++) {
                float p = ex[e] * inv;
                atomicAdd(&sProb[e], p);
                if (p > best) { best = p; sel = e; }   // first-max wins (strict >)
            }
            expert_idx[s] = sel;
            gate_val[s]   = best;
            atomicAdd(&sCnt[sel], 1);
        }
    }
    __syncthreads();
    if (tid < NEXP) { atomicAdd(&prob_sum[tid], sProb[tid]); atomicAdd(&cnt[tid], sCnt[tid]); }
}

// ---------------- scan: per-expert positions + slot table ----------------
#define SCAN_T 256
__global__ void scan_kernel(const int* __restrict__ expert_idx,
                            int* __restrict__ token_for_slot,
                            int S, int cap) {
    __shared__ int sE[SCAN_T];
    __shared__ int base[NEXP];
    int tid = threadIdx.x;
    if (tid < NEXP) base[tid] = 0;
    __syncthreads();
    for (int start = 0; start < S; start += SCAN_T) {
        int t = start + tid;
        int e = (t < S) ? expert_idx[t] : -1;
        sE[tid] = e;
        __syncthreads();
        int rank = 0;
        for (int i = 0; i < tid; i++) rank += (sE[i] == e) ? 1 : 0;
        if (e >= 0) {
            int pos = base[e] + rank;
            if (pos < cap) token_for_slot[e * cap + pos] = t;
        }
        __syncthreads();
        if (tid < NEXP) {
            int c = 0;
            for (int i = 0; i < SCAN_T; i++) c += (sE[i] == tid) ? 1 : 0;
            base[tid] += c;
        }
        __syncthreads();
    }
}

// ---------------- fused gather + grouped GEMM + combine scatter ----------------
// 4 waves per block share one (expert, 64-col) B slice for L0 locality.
// Each wave computes a 32x64 tile of expert_out (2 row-tiles x 4 n-tiles ->
// 8 x v_wmma_f32_16x16x4_f32 per K-step), then scatters rows to token order
// scaled by the gate value.
__global__ __launch_bounds__(128)
void moe_gemm_kernel(const float* __restrict__ features,
                     const float* __restrict__ expert_w,
                     const float* __restrict__ expert_b,
                     const int* __restrict__ token_for_slot,
                     const float* __restrict__ gate_val,
                     float* __restrict__ out, int cap) {
    int groupsPerExp = cap >> 7;                 // 128 rows per block
    int e       = blockIdx.x / groupsPerExp;
    int rowInE  = (blockIdx.x % groupsPerExp) << 7;
    int nBase   = blockIdx.y << 6;               // 64 columns per block
    int tid     = threadIdx.x;
    int wid     = tid >> 5;                      // wave id: 32-row sub-tile
    int lane    = tid & 31;
    int h       = lane >> 4;                     // half-wave: K pair select
    int l16     = lane & 15;

    __shared__ int   sTok[128];
    __shared__ float sGate[128];
    {
        int tk = token_for_slot[e * cap + rowInE + tid];
        sTok[tid]  = tk;
        sGate[tid] = (tk >= 0) ? gate_val[tk] : 0.f;
    }
    __syncthreads();

    int rowBase = wid << 5;                      // this wave's 32 rows in block
    int tok0 = sTok[rowBase + l16];
    int tok1 = sTok[rowBase + 16 + l16];

    // A fragments: lane (h,l16) holds A[row=l16][k+2h], A[row=l16][k+2h+1]
    const float* aPtr0 = features + (size_t)(tok0 < 0 ? 0 : tok0) * MDIM + (h << 1);
    const float* aPtr1 = features + (size_t)(tok1 < 0 ? 0 : tok1) * MDIM + (h << 1);
    // B fragment: lane (h,l16) holds B[k+2h][n], B[k+2h+1][n]
    const float* bPtr = expert_w + (size_t)e * (MDIM * ODIM)
                      + (size_t)(h << 1) * ODIM + nBase + l16;

    v8f zero = {0.f, 0.f, 0.f, 0.f, 0.f, 0.f, 0.f, 0.f};
    v8f acc[2][4] = {{zero, zero, zero, zero}, {zero, zero, zero, zero}};

    for (int k = 0; k < MDIM; k += 4) {
        v2f a0 = *(const v2f*)(aPtr0 + k);
        v2f a1 = *(const v2f*)(aPtr1 + k);
        const float* bk = bPtr + (size_t)k * ODIM;
        if ((k & 15) == 0 && k + 16 < MDIM) {
            // stream-ahead hint for the B slice (global_prefetch_b8)
            __builtin_prefetch(bk + 16 * ODIM, 0, 1);
        }
#pragma unroll
        for (int t = 0; t < 4; t++) {
            v2f b;
            b.x = bk[t * 16];
            b.y = bk[t * 16 + ODIM];
            acc[0][t] = __builtin_amdgcn_wmma_f32_16x16x4_f32(
                false, a0, false, b, (short)0, acc[0][t], false, false);
            acc[1][t] = __builtin_amdgcn_wmma_f32_16x16x4_f32(
                false, a1, false, b, (short)0, acc[1][t], false, false);
        }
    }

    // epilogue: C layout -> lane holds col N=l16, rows M = 8*h + j (j=0..7)
#pragma unroll
    for (int rt = 0; rt < 2; rt++) {
#pragma unroll
        for (int t = 0; t < 4; t++) {
#pragma unroll
            for (int j = 0; j < 8; j++) {
                int m  = rowBase + (rt << 4) + (h << 3) + j;
                int tk = sTok[m];
                if (tk >= 0) {
                    int col = nBase + (t << 4) + l16;
                    out[(size_t)tk * ODIM + col] =
                        sGate[m] * (acc[rt][t][j] + expert_b[e * ODIM + col]);
                }
            }
        }
    }
}

// ---------------- l_aux ----------------
__global__ void laux_kernel(const float* __restrict__ prob_sum,
                            const int* __restrict__ cnt,
                            float* __restrict__ out_laux, int S) {
    if (threadIdx.x == 0 && blockIdx.x == 0) {
        float acc = 0.f;
        float invS = 1.f / (float)S;
        for (int e = 0; e < NEXP; e++)
            acc += (prob_sum[e] * invS) * ((float)cnt[e] * invS);
        *out_laux = acc * (float)(NEXP * NEXP);
    }
}

extern "C" void kernel_launch(void* const* d_in, const int* in_sizes, int n_in,
                              void* d_out, int out_size, void* d_ws, size_t ws_size,
                              hipStream_t stream) {
    const float* features = (const float*)d_in[0];   // [B,T,M] -> [S,M]
    const float* gate_w   = (const float*)d_in[1];   // [M,E]
    const float* expert_w = (const float*)d_in[2];   // [E,M,OUT]
    const float* expert_b = (const float*)d_in[3];   // [E,OUT]
    float* out = (float*)d_out;

    int M   = in_sizes[1] / NEXP;                    // 1024
    int S   = in_sizes[0] / M;                       // 8192
    int cap = S / NEXP;                              // 2048
    int outN = S * ODIM;

    // workspace layout
    char* ws = (char*)d_ws;
    int*   token_for_slot = (int*)ws;                                        // NEXP*cap ints
    int*   expert_idx     = (int*)(ws + sizeof(int) * (size_t)(NEXP * cap));
    float* gateval        = (float*)(ws + sizeof(int) * (size_t)(NEXP * cap + S));
    float* prob_sum       = (float*)(ws + sizeof(int) * (size_t)(NEXP * cap + S)
                                        + sizeof(float) * (size_t)S);
    int*   cnt            = (int*)((char*)prob_sum + sizeof(float) * NEXP);

    zero_out_kernel<<<(outN / 4 + 255) / 256, 256, 0, stream>>>(out, outN / 4);
    init_ws_kernel<<<(NEXP * cap + 255) / 256, 256, 0, stream>>>(token_for_slot,
                                                                 prob_sum, cnt, NEXP * cap);
    gate_kernel<<<(S + 7) / 8, 256, 0, stream>>>(features, gate_w, expert_idx,
                                                 gateval, prob_sum, cnt, S);
    scan_kernel<<<1, SCAN_T, 0, stream>>>(expert_idx, token_for_slot, S, cap);

    dim3 grid(NEXP * cap / 128, ODIM / 64);
    moe_gemm_kernel<<<grid, 128, 0, stream>>>(features, expert_w, expert_b,
                                              token_for_slot, gateval, out, cap);

    laux_kernel<<<1, 1, 0, stream>>>(prob_sum, cnt, out + outN, S);
}